// GlobalAttentionPooling_15109694947988
// MI455X (gfx1250) — compile-verified
//
#include <hip/hip_runtime.h>
#include <hip/hip_bf16.h>
#include <stdint.h>

#define DIMS   256     // feature dim D
#define MAXR   256     // rows per LDS tile (256 rows * 1KB = 256KB LDS)
#define BLOCK  256     // 8 waves of 32

typedef unsigned int u32x4 __attribute__((ext_vector_type(4)));
typedef int          i32x8 __attribute__((ext_vector_type(8)));
typedef int          i32x4 __attribute__((ext_vector_type(4)));
typedef float        v2f   __attribute__((ext_vector_type(2)));
typedef float        v8f   __attribute__((ext_vector_type(8)));

__device__ inline int lower_bound_i32(const int* a, int n, int v) {
  int lo = 0, hi = n;
  while (lo < hi) {
    int mid = (lo + hi) >> 1;
    if (a[mid] < v) lo = mid + 1; else hi = mid;
  }
  return lo;
}

// DMA `rows` rows of 256 f32 from global into LDS at lds_off using the
// Tensor Data Mover.  Issued by wave 0 only; everyone syncs after.
__device__ inline void tdm_load_tile(const float* gsrc, unsigned lds_off, int rows) {
  __syncthreads();                       // previous readers of the tile are done
  if (threadIdx.x < 32) {                // one wave issues the DMA
    unsigned long long ga = (unsigned long long)(uintptr_t)gsrc;
    u32x4 g0; i32x8 g1;
    i32x4 g2 = {0, 0, 0, 0};
    i32x4 g3 = {0, 0, 0, 0};
    i32x8 g4 = {0, 0, 0, 0, 0, 0, 0, 0};
    // ---- D# group 0 (128b): count=1 | lds_addr | global_addr[56:0] | type=2
    g0[0] = 1u;
    g0[1] = lds_off;
    g0[2] = (unsigned)(ga & 0xFFFFFFFFull);
    g0[3] = (unsigned)((ga >> 32) & 0x1FFFFFFull) | 0x80000000u;  // type=2 at [127:126]
    // ---- D# group 1 (256b)
    g1[0] = 0x00020000;                                  // data_size = 2 (4 bytes)
    g1[1] = (int)(((unsigned)DIMS & 0xFFFFu) << 16);     // tensor_dim0 lo16 at [63:48]
    g1[2] = (int)(((unsigned)rows & 0xFFFFu) << 16);     // dim0 hi=0 | tensor_dim1 lo16
    g1[3] = (int)(((unsigned)DIMS & 0xFFFFu) << 16);     // tensor_dim1 hi=0 | tile_dim0=256
    g1[4] = rows & 0xFFFF;                               // tile_dim1 | tile_dim2=0
    g1[5] = DIMS;                                        // tensor_dim0_stride lo32
    g1[6] = 0;                                           // stride hi | dim1_stride lo
    g1[7] = 0;
    __builtin_amdgcn_tensor_load_to_lds(g0, g1, g2, g3, g4, 0);
    __builtin_amdgcn_s_wait_tensorcnt(0);
  }
  __syncthreads();                       // data visible to all waves
}

__global__ __launch_bounds__(BLOCK)
void gap_kernel(const float* __restrict__ x, const int* __restrict__ batch,
                const float* __restrict__ Wg, const float* __restrict__ bg,
                float* __restrict__ out, float* __restrict__ ws_gate, int N) {
  extern __shared__ char smem_raw[];
  float* xtile = (float*)smem_raw;            // MAXR * DIMS floats
  float* sattn = xtile + MAXR * DIMS;         // MAXR floats (gate -> e -> attn)
  float* swv   = sattn + MAXR;                // 8 floats  (per-wave partials)
  float* sred  = swv + 8;                     // 2 floats  (gmax, denom)
  int*   sbnd  = (int*)(sred + 2);            // 2 ints    (segment bounds)

  const int g    = blockIdx.x;
  const int tid  = threadIdx.x;
  const int w    = tid >> 5;
  const int lane = tid & 31;

  if (tid == 0) {
    sbnd[0] = lower_bound_i32(batch, N, g);
    sbnd[1] = lower_bound_i32(batch, N, g + 1);
  }
  __syncthreads();
  const int s = sbnd[0];
  const int n = sbnd[1] - s;

  if (n <= 0) {                                // empty segment -> zeros
    out[(size_t)g * DIMS + tid] = 0.0f;
    return;
  }

  const int nt = (n + MAXR - 1) / MAXR;        // #tiles (==1 in practice)
  const unsigned lds_off = (unsigned)(uintptr_t)xtile;  // flat->LDS offset (addr[31:0])
  const float bg0 = bg[0];

  // Per-lane slice of Wg: lane holds Wg[8*lane .. 8*lane+7]
  const float4 wa = ((const float4*)Wg)[2 * lane];
  const float4 wb = ((const float4*)Wg)[2 * lane + 1];

  // ---------- Phase 1: gates + running max (one wave per row) ----------
  float lmax = -3.402823466e38f;
  for (int t = 0; t < nt; ++t) {
    const int t0   = t * MAXR;
    const int rows = min(MAXR, n - t0);
    tdm_load_tile(x + (size_t)(s + t0) * DIMS, lds_off, rows);
    for (int r = w; r < rows; r += 8) {
      const float4* xr = (const float4*)(xtile + r * DIMS);
      float4 xa = xr[2 * lane], xb = xr[2 * lane + 1];
      float sum = xa.x * wa.x + xa.y * wa.y + xa.z * wa.z + xa.w * wa.w
                + xb.x * wb.x + xb.y * wb.y + xb.z * wb.z + xb.w * wb.w;
      for (int off = 16; off > 0; off >>= 1) sum += __shfl_xor(sum, off, 32);
      float gate = sum + bg0;                  // wave-uniform after reduce
      lmax = fmaxf(lmax, gate);
      if (lane == 0) {
        if (nt == 1) sattn[r] = gate;          // fast path: keep in LDS
        else         ws_gate[s + t0 + r] = gate;  // rare path: spill to global ws
      }
    }
  }
  if (nt > 1) __threadfence();
  __syncthreads();

  // block-wide max
  if (lane == 0) swv[w] = lmax;
  __syncthreads();
  if (tid == 0) {
    float m = swv[0];
    for (int i = 1; i < 8; ++i) m = fmaxf(m, swv[i]);
    sred[0] = m;
  }
  __syncthreads();
  const float gmax = sred[0];

  // ---------- Phase 2: softmax denominator (fast path caches e in sattn) ----
  float part = 0.0f;
  for (int r = tid; r < n; r += BLOCK) {
    float gate = (nt == 1) ? sattn[r] : ws_gate[s + r];
    float e = __expf(gate - gmax);
    part += e;
    if (nt == 1) sattn[r] = e;                 // each r owned by one thread
  }
  for (int off = 16; off > 0; off >>= 1) part += __shfl_xor(part, off, 32);
  if (lane == 0) swv[w] = part;
  __syncthreads();
  if (tid == 0) {
    float d2 = 0.0f;
    for (int i = 0; i < 8; ++i) d2 += swv[i];
    sred[1] = d2;
  }
  __syncthreads();
  const float invd = 1.0f / sred[1];

  // ---------- Phase 3: weighted pooling via V_WMMA_F32_16X16X4_F32 ----------
  // Wave w owns output columns [32w, 32w+32): two 16-wide accumulators.
  // A (16x4 f32): lanes 0-15 VGPR0/1 = K0/K1; lanes 16-31 = K2/K3; every M row
  // carries the same (a0..a3), so D row 0 == desired column partial sums.
  // Tail rows are zero-PADDED (attn=0 and x=0) so the hot loop has no guards
  // and EXEC stays all-ones throughout.
  const int half = lane >> 4;
  const int ln   = lane & 15;
  const int c0   = w * 32, c1 = c0 + 16;
  v8f acc0 = {0, 0, 0, 0, 0, 0, 0, 0};
  v8f acc1 = {0, 0, 0, 0, 0, 0, 0, 0};

  for (int t = 0; t < nt; ++t) {
    const int t0    = t * MAXR;
    const int rows  = min(MAXR, n - t0);
    const int rows4 = (rows + 3) & ~3;         // <= MAXR (MAXR % 4 == 0)
    if (nt > 1) tdm_load_tile(x + (size_t)(s + t0) * DIMS, lds_off, rows);
    // attn for this tile (in-place in sattn) + zero padding
    for (int r = tid; r < rows; r += BLOCK) {
      sattn[r] = (nt == 1) ? sattn[r] * invd
                           : __expf(ws_gate[s + t0 + r] - gmax) * invd;
    }
    for (int r = rows + tid; r < rows4; r += BLOCK) sattn[r] = 0.0f;
    for (int i = tid; i < (rows4 - rows) * DIMS; i += BLOCK)
      xtile[rows * DIMS + i] = 0.0f;           // pad x rows: garbage*0 -> 0, no NaN
    __syncthreads();
    for (int r = 0; r < rows4; r += 4) {
      const int k0 = r + 2 * half;             // K index pair for this lane half
      const int k1 = k0 + 1;
      v2f A, B0, B1;
      A[0]  = sattn[k0];
      A[1]  = sattn[k1];
      B0[0] = xtile[k0 * DIMS + c0 + ln];
      B0[1] = xtile[k1 * DIMS + c0 + ln];
      B1[0] = xtile[k0 * DIMS + c1 + ln];
      B1[1] = xtile[k1 * DIMS + c1 + ln];
      acc0 = __builtin_amdgcn_wmma_f32_16x16x4_f32(false, A, false, B0,
                                                   (short)0, acc0, false, false);
      acc1 = __builtin_amdgcn_wmma_f32_16x16x4_f32(false, A, false, B1,
                                                   (short)0, acc1, false, false);
    }
    __syncthreads();                           // done reading sattn/xtile
  }

  // D-matrix row M=0 lives in C VGPR0, lanes 0-15 (N = lane)
  if (lane < 16) {
    out[(size_t)g * DIMS + c0 + ln] = acc0[0];
    out[(size_t)g * DIMS + c1 + ln] = acc1[0];
  }
}

extern "C" void kernel_launch(void* const* d_in, const int* in_sizes, int n_in,
                              void* d_out, int out_size, void* d_ws, size_t ws_size,
                              hipStream_t stream) {
  const float* x     = (const float*)d_in[0];
  const int*   batch = (const int*)d_in[1];
  const float* Wg    = (const float*)d_in[2];
  const float* bg    = (const float*)d_in[3];
  float*       out   = (float*)d_out;
  float*       ws    = (float*)d_ws;       // gate spill for oversized segments

  const int N = in_sizes[1];               // number of nodes
  const int G = out_size / DIMS;           // number of segments

  const size_t smem = (size_t)MAXR * DIMS * sizeof(float)   // x tile
                    + (MAXR + 8 + 2) * sizeof(float)        // attn + reduce scratch
                    + 2 * sizeof(int);                      // bounds
  (void)hipFuncSetAttribute((const void*)gap_kernel,
                            hipFuncAttributeMaxDynamicSharedMemorySize, (int)smem);
  gap_kernel<<<G, BLOCK, smem, stream>>>(x, batch, Wg, bg, out, ws, N);
}